// ProtoPNetConv_33139967656188
// MI455X (gfx1250) — compile-verified
//
#include <hip/hip_runtime.h>

typedef float v2f __attribute__((ext_vector_type(2)));
typedef float v8f __attribute__((ext_vector_type(8)));

#define BATCH   64
#define SEQ     512
#define ENC     1024
#define NPROT   100
#define KP      4
#define CUT     509      // SEQ - KP + 1
#define PDIM    4096     // KP * ENC (flat prototype length)

#define WAVES   7        // one wave per 16-wide prototype tile (7*16 = 112 >= 100)
#define TPB     (WAVES * 32)
#define CC      64       // channels staged per LDS chunk
#define ROWS    20       // 16 output rows + (KP-1) halo, padded to 20
#define RSTRIDE 21       // odd-ish stride -> fewer LDS bank conflicts
#define ATILE   (CC * RSTRIDE)

// ---------------------------------------------------------------------------
// Kernel 1: x2sum[b,l] = sum_c emb[b,l,c]^2   and   p2[p] = sum proto[p]^2
// One wave32 per reduction row.
// ---------------------------------------------------------------------------
__global__ __launch_bounds__(256) void precompute_kernel(
    const float* __restrict__ emb, const float* __restrict__ proto,
    float* __restrict__ x2sum, float* __restrict__ p2) {
  int wid  = (blockIdx.x * blockDim.x + threadIdx.x) >> 5;
  int lane = threadIdx.x & 31;
  if (wid < BATCH * SEQ) {
    const float* row = emb + (size_t)wid * ENC;
    float s = 0.f;
    for (int i = lane; i < ENC; i += 32) { float v = row[i]; s += v * v; }
    for (int m = 16; m >= 1; m >>= 1) s += __shfl_xor(s, m, 32);
    if (lane == 0) x2sum[wid] = s;
  } else if (wid < BATCH * SEQ + NPROT) {
    int p = wid - BATCH * SEQ;
    const float* row = proto + (size_t)p * PDIM;
    float s = 0.f;
    for (int i = lane; i < PDIM; i += 32) { float v = row[i]; s += v * v; }
    for (int m = 16; m >= 1; m >>= 1) s += __shfl_xor(s, m, 32);
    if (lane == 0) p2[p] = s;
  }
}

// ---------------------------------------------------------------------------
// Async DMA staging: global embedding tile -> LDS (transposed [chan][row]),
// tracked on ASYNCcnt (no VGPR round trip). GVS addressing: saddr = emb base,
// vaddr = 32-bit byte offset; LDS address = low 32 bits of the flat pointer.
// ---------------------------------------------------------------------------
__device__ __forceinline__ void stage_async(const float* __restrict__ embb,
                                            float* ldsbase, int t0, int c0,
                                            int tid) {
  for (int i = tid; i < ROWS * CC; i += TPB) {
    int r = i / CC;
    int c = i - r * CC;
    int lr = t0 + r; if (lr > SEQ - 1) lr = SEQ - 1;   // halo clamp (unused rows)
    unsigned lds_off = (unsigned)(size_t)(ldsbase + (c * RSTRIDE + r));
    unsigned g_off   = (unsigned)((lr * ENC + (c0 + c)) * 4);
    asm volatile("global_load_async_to_lds_b32 %0, %1, %2"
                 :: "v"(lds_off), "v"(g_off), "s"(embb)
                 : "memory");
  }
}

__device__ __forceinline__ void wait_async_all() {
  asm volatile("s_wait_asynccnt 0x0" ::: "memory");
}

// ---------------------------------------------------------------------------
// Kernel 2: main GEMM via V_WMMA_F32_16X16X4_F32 + distance epilogue.
// Grid: (SEQ/16 t-tiles, BATCH). Block: 7 waves, wave w owns p-tile [16w,16w+16).
// One WMMA K-step == one channel c (K-lanes 0..3 are taps j=0..3).
// Double-buffered LDS; next chunk prefetched via async-to-LDS during compute.
// ---------------------------------------------------------------------------
__global__ __launch_bounds__(TPB) void proto_dist_kernel(
    const float* __restrict__ emb, const float* __restrict__ proto,
    const float* __restrict__ x2sum, const float* __restrict__ p2,
    float* __restrict__ distances) {
  __shared__ float alds[2][ATILE];       // [buf][channel][row] transposed tiles

  const int t0   = blockIdx.x * 16;
  const int b    = blockIdx.y;
  const int tid  = threadIdx.x;
  const int wave = tid >> 5;
  const int lane = tid & 31;
  const int half = lane >> 4;            // 0: K=0..1 ; 1: K=2..3
  const int lp   = lane & 15;

  const int p  = wave * 16 + lp;
  const int pc = (p < NPROT) ? p : (NPROT - 1);   // clamp, stores are guarded

  const float* embb = emb + (size_t)b * SEQ * ENC;
  const float* bptr = proto + (size_t)pc * PDIM + half * 2;

  // Prologue: fill buffer 0 with chunk 0.
  stage_async(embb, &alds[0][0], t0, 0, tid);
  wait_async_all();
  __syncthreads();

  v8f acc = {};
  for (int c0 = 0; c0 < ENC; c0 += CC) {
    const int cur = (c0 / CC) & 1;
    // Prefetch next chunk into the other buffer (overlaps with compute below).
    if (c0 + CC < ENC)
      stage_async(embb, &alds[cur ^ 1][0], t0, c0 + CC, tid);

    const float* abase = &alds[cur][0];
#pragma unroll 4
    for (int cc = 0; cc < CC; ++cc) {
      const int c = c0 + cc;
      // A fragment: rows (lp+half*2, lp+half*2+1) at this channel (ISA 16x4 layout)
      const float* ap = abase + (cc * RSTRIDE + lp + half * 2);
      v2f a; a.x = ap[0]; a.y = ap[1];
      // B fragment: proto_flat[p][4c + 2*half .. +1] (contiguous float2)
      const float* bp = bptr + (size_t)c * 4;
      v2f bb; bb.x = bp[0]; bb.y = bp[1];
      acc = __builtin_amdgcn_wmma_f32_16x16x4_f32(false, a, false, bb,
                                                  (short)0, acc, false, false);
    }

    // Own DMA (issued above) complete; all waves done reading `cur`.
    wait_async_all();
    __syncthreads();
  }

  // Epilogue: distances[b,p,t] = sqrt(|x2_patch_sum - 2*xp + p2|)
  const float  pp2  = p2[pc];
  const int    g    = pc / 25;                    // dilation group
  const int    dil  = g + 1;
  const int    padv = (g == 0) ? 0 : (g == 1 ? 2 : (g == 2 ? 3 : 5));
  const float* x2b  = x2sum + b * SEQ;

  if (p < NPROT) {
#pragma unroll
    for (int r = 0; r < 8; ++r) {
      int t = t0 + r + half * 8;                  // C/D layout: M = r + 8*half
      if (t < CUT) {
        float xs = 0.f;
#pragma unroll
        for (int j = 0; j < KP; ++j) {
          int idx = t - padv + j * dil;
          if (idx >= 0 && idx < SEQ) xs += x2b[idx];
        }
        float d = xs - 2.0f * acc[r] + pp2;
        distances[((size_t)(b * NPROT + p)) * CUT + t] = sqrtf(fabsf(d));
      }
    }
  }
}

// ---------------------------------------------------------------------------
// Kernel 3: prototype_distances[b,p] = min_t distances[b,p,t]. Wave per row.
// ---------------------------------------------------------------------------
__global__ __launch_bounds__(256) void min_kernel(
    const float* __restrict__ distances, float* __restrict__ protod) {
  int wid  = (blockIdx.x * blockDim.x + threadIdx.x) >> 5;
  int lane = threadIdx.x & 31;
  if (wid >= BATCH * NPROT) return;
  const float* row = distances + (size_t)wid * CUT;
  float m = 3.402823466e38f;
  for (int t = lane; t < CUT; t += 32) m = fminf(m, row[t]);
  for (int s = 16; s >= 1; s >>= 1) m = fminf(m, __shfl_xor(m, s, 32));
  if (lane == 0) protod[wid] = m;
}

// ---------------------------------------------------------------------------
// Kernel 4: class_out = prototype_distances @ fc_weight.T   (64 x 3)
// ---------------------------------------------------------------------------
__global__ __launch_bounds__(192) void fc_kernel(
    const float* __restrict__ protod, const float* __restrict__ fcw,
    float* __restrict__ out) {
  int i  = threadIdx.x;          // 0..191
  int b  = i / 3;
  int cl = i % 3;
  float s = 0.f;
  for (int pp = 0; pp < NPROT; ++pp)
    s += protod[b * NPROT + pp] * fcw[cl * NPROT + pp];
  out[i] = s;
}

// ---------------------------------------------------------------------------
extern "C" void kernel_launch(void* const* d_in, const int* in_sizes, int n_in,
                              void* d_out, int out_size, void* d_ws, size_t ws_size,
                              hipStream_t stream) {
  const float* emb   = (const float*)d_in[0];   // [64,512,1024]
  const float* proto = (const float*)d_in[1];   // [100,1024,4]
  const float* fcw   = (const float*)d_in[2];   // [3,100]

  float* out       = (float*)d_out;
  float* protod    = out;                                   // [64*100]
  float* distances = out + BATCH * NPROT;                   // [64*100*509]
  float* classout  = distances + (size_t)BATCH * NPROT * CUT;  // [64*3]

  float* x2sum = (float*)d_ws;               // [64*512]
  float* p2    = x2sum + BATCH * SEQ;        // [100]

  {
    int nwaves = BATCH * SEQ + NPROT;
    int blocks = (nwaves * 32 + 255) / 256;
    precompute_kernel<<<blocks, 256, 0, stream>>>(emb, proto, x2sum, p2);
  }
  {
    dim3 grid(SEQ / 16, BATCH);              // 32 x 64 workgroups
    proto_dist_kernel<<<grid, TPB, 0, stream>>>(emb, proto, x2sum, p2, distances);
  }
  {
    int blocks = (BATCH * NPROT * 32 + 255) / 256;
    min_kernel<<<blocks, 256, 0, stream>>>(distances, protod);
  }
  fc_kernel<<<1, 192, 0, stream>>>(protod, fcw, classout);
}